// GraphConv_30468497997919
// MI455X (gfx1250) — compile-verified
//
#include <hip/hip_runtime.h>

#define N_NODES 12288
#define IN_F    128
#define OUT_F   32
#define KSPLIT  2
#define WAVES   8   // waves per block in the streaming GEMM (each wave: one 16-row i-tile)

typedef float v2f __attribute__((ext_vector_type(2)));
typedef float v8f __attribute__((ext_vector_type(8)));

// ---------------------------------------------------------------------------
// Kernel 1: Xw = X @ W   [N,128] x [128,32] -> [N,32]   (tiny, VALU)
// One wave per output row; lane = output column (OUT_F == warpSize == 32).
// ---------------------------------------------------------------------------
__global__ void __launch_bounds__(256) xw_kernel(const float* __restrict__ X,
                                                 const float* __restrict__ W,
                                                 float* __restrict__ Xw) {
    __shared__ float sW[IN_F * OUT_F];
    for (int t = threadIdx.x; t < IN_F * OUT_F; t += blockDim.x) sW[t] = W[t];
    __syncthreads();

    const int wave = threadIdx.x >> 5;
    const int lane = threadIdx.x & 31;
    const int row  = blockIdx.x * 8 + wave;
    if (row >= N_NODES) return;

    const float* xr = X + (size_t)row * IN_F;
    float acc = 0.0f;
#pragma unroll 8
    for (int k = 0; k < IN_F; ++k)
        acc = fmaf(xr[k], sW[k * OUT_F + lane], acc);

    Xw[(size_t)row * OUT_F + lane] = acc;
}

// ---------------------------------------------------------------------------
// Kernel 2: streaming GEMM  P_s = adj(Krange)^T @ Xw(Krange)  via
// V_WMMA_F32_16X16X4_F32, fused with per-column sums of adj (degree).
//
// A-tile (16x4 f32) per ISA layout: lanes 0-15 hold M=lane; v0 = K0 (half 0)
// or K2 (half 1), v1 = K1 / K3.  B-tile (4x16) mirrors: v0 = {K0|K2} with
// N = lane&15, v1 = {K1|K3}.  C/D: v8f, element r -> row r + 8*(lane>>4).
// ---------------------------------------------------------------------------
__global__ void __launch_bounds__(32 * WAVES, 1) spmm_kernel(
    const float* __restrict__ adj, const float* __restrict__ Xw,
    float* __restrict__ Ppart, float* __restrict__ csPart) {

    const int lane = threadIdx.x & 31;
    const int wave = threadIdx.x >> 5;
    const int half = lane >> 4;   // 0: K0/K1 rows, 1: K2/K3 rows
    const int lm   = lane & 15;

    const int tilesPerSplit = N_NODES / 16 / WAVES;          // 96
    const int s  = blockIdx.x / tilesPerSplit;               // K-split id
    const int tb = blockIdx.x % tilesPerSplit;
    const int i0 = tb * (16 * WAVES) + wave * 16;            // this wave's i-tile

    const int Kseg = N_NODES / KSPLIT;                       // 6144
    const int kbeg = s * Kseg;

    v8f c0 = {};   // j in [0,16)
    v8f c1 = {};   // j in [16,32)
    float sacc = 0.0f;  // partial column sum of adj for column i0+lm (split over halves)

    const float* aptr  = adj + (size_t)(kbeg + 2 * half) * N_NODES + (i0 + lm);
    const float* bptr0 = Xw  + (size_t)(kbeg + 2 * half) * OUT_F + lm;
    const float* bptr1 = bptr0 + 16;

#pragma unroll 8
    for (int kk = 0; kk < Kseg; kk += 4) {
        // adj is streamed exactly once -> nontemporal; Xw stays hot in L2.
        float a0 = __builtin_nontemporal_load(aptr);            // row k+2h
        float a1 = __builtin_nontemporal_load(aptr + N_NODES);  // row k+2h+1
        float b00 = bptr0[0];
        float b01 = bptr0[OUT_F];
        float b10 = bptr1[0];
        float b11 = bptr1[OUT_F];
        aptr  += 4 * (size_t)N_NODES;
        bptr0 += 4 * OUT_F;
        bptr1 += 4 * OUT_F;

        sacc += a0 + a1;                    // degree accumulation for free

        v2f a  = { a0,  a1  };
        v2f b0 = { b00, b01 };
        v2f b1 = { b10, b11 };
        c0 = __builtin_amdgcn_wmma_f32_16x16x4_f32(false, a, false, b0,
                                                   (short)0, c0, false, false);
        c1 = __builtin_amdgcn_wmma_f32_16x16x4_f32(false, a, false, b1,
                                                   (short)0, c1, false, false);
    }

    // Fold the two half-wave column-sum partials: col i0+l = lane l + lane l+16.
    float other = __shfl_down(sacc, 16, 32);
    if (lane < 16)
        csPart[(size_t)s * N_NODES + i0 + lane] = sacc + other;

    // Store partial P tile.
    float* prow = Ppart + ((size_t)s * N_NODES + i0) * OUT_F;
#pragma unroll
    for (int r = 0; r < 8; ++r) {
        const int irow = r + 8 * half;
        prow[(size_t)irow * OUT_F + lm]      = c0[r];
        prow[(size_t)irow * OUT_F + 16 + lm] = c1[r];
    }
}

// ---------------------------------------------------------------------------
// Kernel 3: reduce K-split partials + diagonal fix + degree normalization.
// out[i,j] = dinv(i) * ( P[i,j] + (1 - adj[i,i]) * Xw[i,j] )
// deg(i)   = colsum(i) - adj[i,i] + 1
// ---------------------------------------------------------------------------
__global__ void __launch_bounds__(256) epilogue_kernel(
    const float* __restrict__ adj, const float* __restrict__ Xw,
    const float* __restrict__ Ppart, const float* __restrict__ csPart,
    float* __restrict__ out) {

    const int idx = blockIdx.x * blockDim.x + threadIdx.x;
    if (idx >= N_NODES * OUT_F) return;
    const int i = idx >> 5;
    const int j = idx & 31;

    float pre = 0.0f, cs = 0.0f;
#pragma unroll
    for (int s = 0; s < KSPLIT; ++s) {
        pre += Ppart[((size_t)s * N_NODES + i) * OUT_F + j];
        cs  += csPart[(size_t)s * N_NODES + i];
    }
    const float diag = adj[(size_t)i * N_NODES + i];
    const float xw   = Xw[(size_t)i * OUT_F + j];
    const float deg  = cs - diag + 1.0f;
    const float dinv = (deg == 0.0f) ? 0.0f : 1.0f / deg;
    out[idx] = dinv * (pre + (1.0f - diag) * xw);
}

// ---------------------------------------------------------------------------
extern "C" void kernel_launch(void* const* d_in, const int* in_sizes, int n_in,
                              void* d_out, int out_size, void* d_ws, size_t ws_size,
                              hipStream_t stream) {
    const float* X   = (const float*)d_in[0];   // [N, 128]
    const float* adj = (const float*)d_in[1];   // [N, N]
    const float* W   = (const float*)d_in[2];   // [128, 32]
    // d_in[3] = norm scalar (unused; reference hardcodes the normalized path)

    float* ws     = (float*)d_ws;
    float* Xw     = ws;                                          // N*32
    float* Ppart  = Xw + (size_t)N_NODES * OUT_F;                // KSPLIT*N*32
    float* csPart = Ppart + (size_t)KSPLIT * N_NODES * OUT_F;    // KSPLIT*N
    float* out    = (float*)d_out;

    xw_kernel<<<N_NODES / 8, 256, 0, stream>>>(X, W, Xw);
    spmm_kernel<<<(N_NODES / 16 / WAVES) * KSPLIT, 32 * WAVES, 0, stream>>>(
        adj, Xw, Ppart, csPart);
    epilogue_kernel<<<(N_NODES * OUT_F) / 256, 256, 0, stream>>>(
        adj, Xw, Ppart, csPart, out);
}